// Fusedmax_84155589197989
// MI455X (gfx1250) — compile-verified
//
#include <hip/hip_runtime.h>
#include <hip/hip_bf16.h>
#include <stdint.h>

typedef __attribute__((ext_vector_type(2))) float v2f;
typedef __attribute__((ext_vector_type(8))) float v8f;

#define TVN 512   // row length (reference: 2048 x 512)

// ---------------------------------------------------------------------------
// Kernel 1: exact 1-D TV prox (Condat 2013), one thread per row.
// Sequential & branchy by nature; state lives in registers, row data hits L2.
// 32-thread (1-wave) blocks: 64 workgroups -> max WGP spread for this
// latency-bound phase. Writes rows into d_out (overwritten by kernel 2).
// ---------------------------------------------------------------------------
__global__ __launch_bounds__(32) void tv1d_rows_kernel(const float* __restrict__ x,
                                                       float* __restrict__ y,
                                                       int rows) {
    const float lam = 1.0f;  // ALPHA
    int row = blockIdx.x * blockDim.x + threadIdx.x;
    if (row >= rows) return;
    const float* __restrict__ xr = x + (size_t)row * TVN;
    float* __restrict__ yr = y + (size_t)row * TVN;
    __builtin_prefetch(xr, 0, 0);   // global_prefetch_b8: pull row toward WGP

    int k = 0, k0 = 0, km = 0, kp = 0;
    float vmin = xr[0] - lam, vmax = xr[0] + lam;
    float umin = lam, umax = -lam;

    for (;;) {
        if (k >= TVN - 1) {                       // terminal cases
            if (umin < 0.0f) {                    // lower constraint violated
                for (int i = k0; i <= km; ++i) yr[i] = vmin;
                int k2 = km + 1;
                k = k0 = km = k2;
                int kc = (k2 < TVN - 1) ? k2 : (TVN - 1);   // clamped read
                float xk = xr[kc];
                umin = lam;
                umax = xk + lam - vmax;           // vmax unchanged
                vmin = xk;
            } else if (umax > 0.0f) {             // upper constraint violated
                for (int i = k0; i <= kp; ++i) yr[i] = vmax;
                int k2 = kp + 1;
                k = k0 = kp = k2;
                int kc = (k2 < TVN - 1) ? k2 : (TVN - 1);
                float xk = xr[kc];
                umin = xk - lam - vmin;           // vmin unchanged
                umax = -lam;
                vmax = xk;
            } else {                              // done: flush last segment
                float v = vmin + umin / (float)(k - k0 + 1);
                for (int i = k0; i < TVN; ++i) yr[i] = v;
                break;
            }
        } else {
            float xk1 = xr[k + 1];
            if (xk1 + umin < vmin - lam) {        // negative jump
                for (int i = k0; i <= km; ++i) yr[i] = vmin;
                int k2 = km + 1;
                k = k0 = km = kp = k2;
                int kc = (k2 < TVN - 1) ? k2 : (TVN - 1);
                float xk = xr[kc];
                vmin = xk; vmax = xk + 2.0f * lam;
                umin = lam; umax = -lam;
            } else if (xk1 + umax > vmax + lam) { // positive jump
                for (int i = k0; i <= kp; ++i) yr[i] = vmax;
                int k2 = kp + 1;
                k = k0 = km = kp = k2;
                int kc = (k2 < TVN - 1) ? k2 : (TVN - 1);
                float xk = xr[kc];
                vmin = xk - 2.0f * lam; vmax = xk;
                umin = lam; umax = -lam;
            } else {                              // no jump: extend segment
                int k2 = k + 1;
                float u1 = umin + xk1 - vmin;
                float u2 = umax + xk1 - vmax;
                float denom = (float)(k2 - k0 + 1);
                if (u1 >= lam) { vmin += (u1 - lam) / denom; umin = lam;  km = k2; }
                else           { umin = u1; }
                if (u2 <= -lam){ vmax += (u2 + lam) / denom; umax = -lam; kp = k2; }
                else           { umax = u2; }
                k = k2;
            }
        }
    }
}

// ---------------------------------------------------------------------------
// Kernel 2: row-wise sparsemax. One 256-thread block (8 wave32s) per row.
//  - async-LDS row staging (gfx1250 GLOBAL_LOAD_ASYNC_TO_LDS_B128 + ASYNCcnt)
//  - bitonic sort descending (512 elems, 256 threads); workgroup barriers only
//    on cross-wave stages (j>=32) or into one — wave-local stages use a
//    wave scheduling fence + per-wave in-order DS guarantees (wave32: thread t
//    owns elements t and t+256, so j<=16 pairs never leave a wave).
//  - chunked inclusive prefix sum via V_WMMA_F32_16X16X4_F32:
//      D = L(16x16, lower-tri ones) * X(16 chunks as columns), 4 accumulating
//      K=4 WMMAs per group of 16 chunks.
//  - support count -> tau -> out = max(y - tau, 0)
// ---------------------------------------------------------------------------
__global__ __launch_bounds__(256) void fused_sparsemax_kernel(float* __restrict__ y,
                                                              int rows) {
    constexpr int N = TVN;
    __shared__ __align__(16) float rowv[N];   // original TV output (unsorted)
    __shared__ float srt[N];                  // sort buffer (descending)
    __shared__ float pre[N];                  // within-chunk inclusive prefix
    __shared__ float coff[N / 16];            // exclusive chunk offsets
    __shared__ int   kz_s;
    __shared__ float tau_s;

    const int tid = threadIdx.x;
    float* __restrict__ yr = y + (size_t)blockIdx.x * N;

    if (tid == 0) kz_s = 0;

    // --- async row load: 128 lanes x 16 bytes -> LDS ---
    if (tid < N / 4) {
        uint32_t lds_off = (uint32_t)(uintptr_t)&rowv[tid * 4];  // low 32b = LDS byte addr
        uint32_t voff    = (uint32_t)(tid * 16);
        asm volatile("global_load_async_to_lds_b128 %0, %1, %2"
                     :: "v"(lds_off), "v"(voff), "s"(yr) : "memory");
    }
    asm volatile("s_wait_asynccnt 0" ::: "memory");
    __syncthreads();

    for (int i = tid; i < N; i += 256) srt[i] = rowv[i];
    __syncthreads();

    // --- bitonic sort, descending ---
    for (int k = 2; k <= N; k <<= 1) {
        for (int j = k >> 1; j > 0; j >>= 1) {
            for (int t = tid; t < N; t += 256) {
                int ixj = t ^ j;
                if (ixj > t) {
                    float a = srt[t], b = srt[ixj];
                    bool up = ((t & k) == 0);
                    if (up ? (a < b) : (a > b)) { srt[t] = b; srt[ixj] = a; }
                }
            }
            // Full barrier only when this stage or the next crosses waves.
            // Next stage distance: j/2 if j>1, else k (first stage of 2k).
            if ((j >= 32) || (j == 1 && k >= 32)) {
                __syncthreads();
            } else {
                __builtin_amdgcn_wave_barrier();   // compiler fence; DS is
            }                                      // in-order within a wave
        }
    }
    // note: the (k=N, j=1) stage above ends with a full __syncthreads()

    // --- WMMA chunked prefix scan (all waves execute, EXEC all-ones;
    //     redundant identical LDS stores across waves are benign) ---
    {
        const int lane  = tid & 31;
        const int m     = lane & 15;               // M of A/D and N (chunk) of B/D
        const int kb    = (lane < 16) ? 0 : 2;     // K pair base for A/B VGPRs
        const int hi8   = (lane < 16) ? 0 : 8;     // M offset for D VGPRs
        #pragma unroll
        for (int g = 0; g < 2; ++g) {
            const int base = g * 256;              // 16 chunks of 16 per group
            v8f acc = {0.f, 0.f, 0.f, 0.f, 0.f, 0.f, 0.f, 0.f};
            #pragma unroll
            for (int j = 0; j < 4; ++j) {
                v2f a, b;
                // A = columns [4j .. 4j+3] of lower-triangular ones L
                a[0] = (4 * j + kb     <= m) ? 1.0f : 0.0f;
                a[1] = (4 * j + kb + 1 <= m) ? 1.0f : 0.0f;
                // B = rows [4j .. 4j+3] of X, chunk m as column
                b[0] = srt[base + m * 16 + 4 * j + kb];
                b[1] = srt[base + m * 16 + 4 * j + kb + 1];
                acc = __builtin_amdgcn_wmma_f32_16x16x4_f32(
                        false, a, false, b, (short)0, acc, false, false);
            }
            #pragma unroll
            for (int v = 0; v < 8; ++v)
                pre[base + m * 16 + v + hi8] = acc[v];   // D[M][n] = prefix(chunk n, M)
        }
    }
    __syncthreads();

    // --- carry scan over 32 chunk totals ---
    if (tid == 0) {
        float s = 0.0f;
        #pragma unroll
        for (int c = 0; c < N / 16; ++c) { coff[c] = s; s += pre[c * 16 + 15]; }
    }
    __syncthreads();

    // --- support size k_z (support is a prefix of the sorted row) ---
    int local = 0;
    for (int i = tid; i < N; i += 256) {
        float cum = pre[i] + coff[i >> 4];
        if (1.0f + (float)(i + 1) * srt[i] > cum) local++;
    }
    atomicAdd(&kz_s, local);
    __syncthreads();

    if (tid == 0) {
        int K = kz_s; if (K < 1) K = 1;
        float cumK = pre[K - 1] + coff[(K - 1) >> 4];
        tau_s = (cumK - 1.0f) / (float)K;
    }
    __syncthreads();

    const float tau = tau_s;
    for (int i = tid; i < N; i += 256)
        yr[i] = fmaxf(rowv[i] - tau, 0.0f);
}

// ---------------------------------------------------------------------------
extern "C" void kernel_launch(void* const* d_in, const int* in_sizes, int n_in,
                              void* d_out, int out_size, void* d_ws, size_t ws_size,
                              hipStream_t stream) {
    const float* x = (const float*)d_in[0];
    float* out = (float*)d_out;
    int rows = in_sizes[0] / TVN;   // 2048 for the reference shapes

    tv1d_rows_kernel<<<(rows + 31) / 32, 32, 0, stream>>>(x, out, rows);
    fused_sparsemax_kernel<<<rows, 256, 0, stream>>>(out, rows);
}